// TrafficGNN_12893491822880
// MI455X (gfx1250) — compile-verified
//
#include <hip/hip_runtime.h>
#include <math.h>

typedef __attribute__((ext_vector_type(2))) float v2f;
typedef __attribute__((ext_vector_type(8))) float v8f;

#define NOUT 128  // hidden/output feature width (fixed by the model)

// ---------------- degree / normalization ----------------
__global__ void k_fill1(float* __restrict__ p, int n) {
  int i = blockIdx.x * blockDim.x + threadIdx.x;
  if (i < n) p[i] = 1.0f;  // self-loop contributes 1 to every degree
}

__global__ void k_deg_edges(const int* __restrict__ dst, float* __restrict__ deg, int E) {
  int i = blockIdx.x * blockDim.x + threadIdx.x;
  if (i < E) atomicAdd(&deg[dst[i]], 1.0f);
}

__global__ void k_rsqrt_inplace(float* __restrict__ p, int n) {
  int i = blockIdx.x * blockDim.x + threadIdx.x;
  if (i < n) { float d = p[i]; p[i] = (d > 0.0f) ? rsqrtf(d) : 0.0f; }
}

// ---------------- aggregation ----------------
// agg[i,f] = xw[i,f] * dinv[i]^2   (self-loop term, also zero-inits the buffer)
__global__ void k_selfloop(const float* __restrict__ xw, const float* __restrict__ dinv,
                           float* __restrict__ agg, int total) {
  int i = blockIdx.x * blockDim.x + threadIdx.x;
  if (i < total) {
    float di = dinv[i >> 7];  // 128 features per node
    agg[i] = xw[i] * di * di;
  }
}

// one wave32 per edge: 4 floats per lane covers the 128-float feature row
__global__ void k_edge_scatter(const float* __restrict__ xw, const float* __restrict__ dinv,
                               const int* __restrict__ src, const int* __restrict__ dst,
                               float* __restrict__ agg, int E) {
  int e = blockIdx.x * 8 + (threadIdx.x >> 5);
  if (e >= E) return;
  int lane = threadIdx.x & 31;
  int s = src[e], d = dst[e];
  float w = dinv[s] * dinv[d];
  const float4* xs = (const float4*)(xw + (size_t)s * NOUT);
  float4 v = xs[lane];
  float* o = agg + (size_t)d * NOUT + lane * 4;
  atomicAdd(o + 0, v.x * w);
  atomicAdd(o + 1, v.y * w);
  atomicAdd(o + 2, v.z * w);
  atomicAdd(o + 3, v.w * w);
}

__global__ void k_bias_relu(float* __restrict__ h, const float* __restrict__ bias, int total) {
  int i = blockIdx.x * blockDim.x + threadIdx.x;
  if (i < total) h[i] = fmaxf(h[i] + bias[i & (NOUT - 1)], 0.0f);
}

// ---------------- WMMA f32 GEMM: C[M,128] (+)= A[M,K] @ W[K,128] ----------------
// flags: bit0 = accumulate into C, bit1 = add bias[col], bit2 = relu
// 8 waves per block: wave w -> column tile w (8*16 = 128 cols); block b -> 16-row stripe.
__global__ __launch_bounds__(256) void k_gemm_wmma(
    const float* __restrict__ A, const float* __restrict__ W,
    const float* __restrict__ bias, float* __restrict__ C,
    int M, int K, int flags) {
  const int lane = threadIdx.x & 31;
  const int wave = threadIdx.x >> 5;
  const int half = lane >> 4;   // 0: lanes 0-15, 1: lanes 16-31
  const int r    = lane & 15;
  const int rowBase = blockIdx.x * 16;
  const int colBase = wave * 16;
  if (rowBase >= M) return;     // block-uniform; EXEC stays all-1s for WMMA

  v8f acc = {};
  // A 16x4 f32 layout: lane half 0 holds K={k0,k0+1}, half 1 holds K={k0+2,k0+3}, row = r
  const float* arow = A + (size_t)(rowBase + r) * K + 2 * half;
  // B 4x16 f32 layout (mirrored): VGPR j = row K=k0+2*half+j, col = colBase + r
  const float* wcol = W + (size_t)(2 * half) * NOUT + colBase + r;
  for (int k0 = 0; k0 < K; k0 += 4) {
    v2f a; a[0] = arow[k0];                 a[1] = arow[k0 + 1];
    v2f b; b[0] = wcol[(size_t)k0 * NOUT];  b[1] = wcol[(size_t)(k0 + 1) * NOUT];
    acc = __builtin_amdgcn_wmma_f32_16x16x4_f32(
        /*neg_a=*/false, a, /*neg_b=*/false, b,
        /*c_mod=*/(short)0, acc, /*reuse_a=*/false, /*reuse_b=*/false);
  }

  float bv = (flags & 2) ? bias[colBase + r] : 0.0f;
  // C/D 16x16 f32 layout: VGPR i -> row = rowBase + half*8 + i, col = colBase + r
#pragma unroll
  for (int i = 0; i < 8; ++i) {
    int row = rowBase + half * 8 + i;
    size_t idx = (size_t)row * NOUT + colBase + r;
    float v = acc[i] + bv;
    if (flags & 1) v += C[idx];
    if (flags & 4) v = fmaxf(v, 0.0f);
    C[idx] = v;
  }
}

// ---------------- host ----------------
extern "C" void kernel_launch(void* const* d_in, const int* in_sizes, int n_in,
                              void* d_out, int out_size, void* d_ws, size_t ws_size,
                              hipStream_t stream) {
  const float* x  = (const float*)d_in[0];
  const int*   ei = (const int*)d_in[1];
  const float* W1 = (const float*)d_in[2];
  const float* b1 = (const float*)d_in[3];
  const float* W2 = (const float*)d_in[4];
  const float* b2 = (const float*)d_in[5];
  const float* Ws = (const float*)d_in[6];
  const float* bs = (const float*)d_in[7];
  const float* Wf = (const float*)d_in[8];
  const float* bf = (const float*)d_in[9];
  float* out = (float*)d_out;

  const int IN = 256;
  const int n  = in_sizes[0] / IN;   // 50000 (multiple of 16)
  const int E  = in_sizes[1] / 2;    // 1.6M
  const int* src = ei;
  const int* dst = ei + E;

  float* ws   = (float*)d_ws;
  float* dinv = ws;                        // n floats (deg, then rsqrt in place)
  float* xw   = ws + n;                    // n*128 (xw1, then xw2)
  float* agg  = xw + (size_t)n * NOUT;     // n*128 (agg1/h1, then agg2/h2)

  const int tot  = n * NOUT;
  dim3 B(256);
  const int gn   = (n + 255) / 256;
  const int gE   = (E + 255) / 256;
  const int gTot = (tot + 255) / 256;
  const int gEw  = (E + 7) / 8;            // 8 waves (edges) per 256-thread block
  const int gM   = n / 16;                 // 3125 row tiles

  // normalization: deg = 1 + indegree; dinv = rsqrt(deg)
  k_fill1        <<<gn,  B, 0, stream>>>(dinv, n);
  k_deg_edges    <<<gE,  B, 0, stream>>>(dst, dinv, E);
  k_rsqrt_inplace<<<gn,  B, 0, stream>>>(dinv, n);

  // layer 1: h1 = relu(Ahat @ (x@W1) + b1)
  k_gemm_wmma    <<<gM,  B, 0, stream>>>(x, W1, nullptr, xw, n, IN, 0);
  k_selfloop     <<<gTot,B, 0, stream>>>(xw, dinv, agg, tot);
  k_edge_scatter <<<gEw, B, 0, stream>>>(xw, dinv, src, dst, agg, E);
  k_bias_relu    <<<gTot,B, 0, stream>>>(agg, b1, tot);

  // layer 2: h2 = relu(Ahat @ (h1@W2) + b2)
  k_gemm_wmma    <<<gM,  B, 0, stream>>>(agg, W2, nullptr, xw, n, NOUT, 0);
  k_selfloop     <<<gTot,B, 0, stream>>>(xw, dinv, agg, tot);
  k_edge_scatter <<<gEw, B, 0, stream>>>(xw, dinv, src, dst, agg, E);
  k_bias_relu    <<<gTot,B, 0, stream>>>(agg, b2, tot);

  // out = h2@Wf + bf  (overwrite poison), then  out += x@Ws + bs
  k_gemm_wmma    <<<gM,  B, 0, stream>>>(agg, Wf, bf, out, n, NOUT, 2);
  k_gemm_wmma    <<<gM,  B, 0, stream>>>(x,  Ws, bs, out, n, IN,  1 | 2);
}